// HBVMulET_33758442946685
// MI455X (gfx1250) — compile-verified
//
#include <hip/hip_runtime.h>
#include <stdint.h>

#define NSTEP 730
#define NGRID 2000
#define PRECS 1e-5f

typedef uint32_t u32x4 __attribute__((ext_vector_type(4)));
typedef uint32_t u32x8 __attribute__((ext_vector_type(8)));

// fast x^y via native transcendentals (v_log_f32 / v_exp_f32); base is always > 0 here
__device__ __forceinline__ float pw(float a, float b) {
    return __builtin_amdgcn_exp2f(b * __builtin_amdgcn_logf(a));
}

__global__ __launch_bounds__(256) void hbv_scan_kernel(
    const float* __restrict__ x,      // (730, 2000, 3)
    const float* __restrict__ par,    // (2000, 13, 16)
    float* __restrict__ out)          // (730, 2000)
{
    constexpr int TC  = 16;                      // timesteps per staged chunk
    constexpr int CE  = TC * 48;                 // 768 floats = 3 KB per chunk
    constexpr int NCH = (NSTEP + TC - 1) / TC;   // 46 chunks

    __shared__ float lx[2][CE];                  // double-buffered forcing (TDM dest)

    const int tid   = threadIdx.x;
    const int m     = tid & 15;                  // mu lane
    const int gl    = tid >> 4;                  // local grid 0..15
    const int gbase = blockIdx.x * 16;
    const int g     = gbase + gl;

    // ---- parameters: load once, de-normalize, keep in registers ----
    constexpr float LO[13] = {1.f,50.f,0.05f,0.01f,0.001f,0.2f,0.f,0.f,-2.5f,0.5f,0.f,0.f,0.3f};
    constexpr float HI[13] = {6.f,1000.f,0.9f,0.5f,0.2f,1.f,10.f,100.f,2.5f,10.f,0.1f,0.2f,5.f};
    float P[13];
    const float* pp = par + (size_t)g * 13 * 16 + m;   // coalesced over m
#pragma unroll
    for (int i = 0; i < 13; ++i) P[i] = LO[i] + pp[i * 16] * (HI[i] - LO[i]);
    const float pBETA = P[0],  pFC   = P[1],  pK0    = P[2],  pK1  = P[3];
    const float pK2   = P[4],  pLP   = P[5],  pPERC  = P[6],  pUZL = P[7];
    const float pTT   = P[8],  pCFMAX= P[9],  pCFR   = P[10], pCWH = P[11];
    const float pBETAET = P[12];
    const float invFC   = 1.f / pFC;
    const float invLPFC = 1.f / (pLP * pFC);
    const float rCF     = pCFR * pCFMAX;

    float SP = 0.001f, MW = 0.001f, SM = 0.001f, SUZ = 0.001f, SLZ = 0.001f;

    // LDS byte offsets of the two staging buffers (low 32 bits of generic addr)
    const uint32_t lbase0 = (uint32_t)(uintptr_t)(&lx[0][0]);
    const uint32_t lbase1 = (uint32_t)(uintptr_t)(&lx[1][0]);

    // ---- Tensor Data Mover: one descriptor copies a whole 16x48-f32 tile ----
    // 2-D tensor view of x rooted at the tile origin: row = one timestep's
    // 48 contiguous floats (16 grids x 3), row stride = NGRID*3 = 6000 floats.
    // tensor_dim1 = rows remaining => TDM zero-fills OOB rows of the last chunk.
    auto tdm_issue = [&](int c, int buf) {
        const int t0   = c * TC;
        const uint32_t rows = (uint32_t)min(TC, NSTEP - t0);
        const uint64_t ga = (uint64_t)(uintptr_t)x
                          + 4ull * (uint32_t)((t0 * NGRID + gbase) * 3);
        u32x4 g0;
        g0.x = 1u;                                            // count=1, load desc valid
        g0.y = buf ? lbase1 : lbase0;                         // lds_addr (bytes)
        g0.z = (uint32_t)ga;                                  // global_addr[31:0]
        g0.w = ((uint32_t)(ga >> 32) & 0x01FFFFFFu)           // global_addr[56:32]
             | 0x80000000u;                                   // type=2 ("image")
        u32x8 g1;
        g1[0] = 0x00020000u;                                  // data_size=2 (4B), mask=0
        g1[1] = 48u << 16;                                    // tensor_dim0[15:0]=48
        g1[2] = (rows << 16);                                 // dim0[31:16]=0 | dim1[15:0]=rows
        g1[3] = (48u << 16);                                  // dim1[31:16]=0 | tile_dim0=48
        g1[4] = (uint32_t)TC;                                 // tile_dim1=16, tile_dim2=0
        g1[5] = 6000u;                                        // tensor_dim0_stride[31:0]
        g1[6] = 0u;                                           // stride[47:32]=0, dim1_stride lo=0
        g1[7] = 0u;
        asm volatile("tensor_load_to_lds %0, %1" :: "s"(g0), "s"(g1) : "memory");
    };

    if (tid < 32) tdm_issue(0, 0);                            // wave 0 drives the DMA
    for (int c = 0; c < NCH; ++c) {
        const int buf = c & 1;
        if (tid < 32 && c + 1 < NCH) tdm_issue(c + 1, buf ^ 1);   // prefetch next chunk
        if (c + 1 < NCH) __builtin_amdgcn_s_wait_tensorcnt(1);    // chunk c landed (in-order)
        else             __builtin_amdgcn_s_wait_tensorcnt(0);
        __syncthreads();                                          // publish LDS to all waves

        const int t0   = c * TC;
        const int tlim = min(TC, NSTEP - t0);
        const float* row = &lx[buf][gl * 3];

        for (int tt = 0; tt < tlim; ++tt) {
            float Pt = row[tt * 48 + 0];          // LDS broadcast to 16 mu lanes
            float Tt = row[tt * 48 + 1];
            float Et = row[tt * 48 + 2];

            float rain = (Tt >= pTT) ? Pt : 0.f;
            float snow = Pt - rain;
            SP += snow;
            float melt = fminf(fmaxf(pCFMAX * (Tt - pTT), 0.f), SP);
            MW += melt; SP -= melt;
            float refr = fminf(fmaxf(rCF * (pTT - Tt), 0.f), MW);
            SP += refr; MW -= refr;
            float tosoil = fmaxf(MW - pCWH * SP, 0.f);
            MW -= tosoil;
            float sw   = fminf(fmaxf(pw(SM * invFC, pBETA), 0.f), 1.f);
            float rech = (rain + tosoil) * sw;
            SM += rain + tosoil - rech;
            float exc = fmaxf(SM - pFC, 0.f);
            SM -= exc;
            float ef = fminf(fmaxf(pw(SM * invLPFC, pBETAET), 0.f), 1.f);
            float ET = fminf(SM, Et * ef);
            SM = fmaxf(SM - ET, PRECS);
            SUZ += rech + exc;
            float PERC = fminf(SUZ, pPERC);
            SUZ -= PERC;
            float Q0 = pK0 * fmaxf(SUZ - pUZL, 0.f);
            SUZ -= Q0;
            float Q1 = pK1 * SUZ;
            SUZ -= Q1;
            SLZ += PERC;
            float Q2 = pK2 * SLZ;
            SLZ -= Q2;

            // mean over 16 mu lanes (off the recurrence's critical path:
            // next state never depends on q, so this overlaps iteration tt+1)
            float q = Q0 + Q1 + Q2;
            q += __shfl_xor(q, 8, 32);
            q += __shfl_xor(q, 4, 32);
            q += __shfl_xor(q, 2, 32);
            q += __shfl_xor(q, 1, 32);
            if (m == 0) out[(t0 + tt) * NGRID + g] = q * (1.0f / 16.0f);
        }
        __syncthreads();   // all reads of lx[buf] done before TDM reuses it next iter
    }
}

extern "C" void kernel_launch(void* const* d_in, const int* in_sizes, int n_in,
                              void* d_out, int out_size, void* d_ws, size_t ws_size,
                              hipStream_t stream) {
    const float* x   = (const float*)d_in[0];   // (730, 2000, 3) f32
    const float* par = (const float*)d_in[1];   // (2000, 13, 16) f32
    float* out = (float*)d_out;                 // (730, 2000, 1) f32
    hipLaunchKernelGGL(hbv_scan_kernel, dim3(NGRID / 16), dim3(256), 0, stream,
                       x, par, out);
}